// FeaturePropagation_85066122265182
// MI455X (gfx1250) — compile-verified
//
#include <hip/hip_runtime.h>

// ---------------------------------------------------------------------------
// FeaturePropagation (PointNet++): 3-NN interp + concat + (1x1conv+BN+ReLU)x2
// Target: MI455X / gfx1250, wave32.
//   - WMMA f32 16x16x4 for both fp32 GEMMs (full reference precision)
//   - GLOBAL_LOAD_ASYNC_TO_LDS_B32 (ASYNCcnt) for the pure-copy skip concat
//   - global_prefetch for weight rows during the NN-scan phase
// ---------------------------------------------------------------------------

typedef __attribute__((ext_vector_type(2))) float v2f;
typedef __attribute__((ext_vector_type(8))) float v8f;

#define B_    8
#define N_    8192
#define S_    2048
#define D1_   128
#define D2_   256
#define CIN_  384      // D2 + D1
#define C0_   256      // conv0 out
#define C1_   128      // conv1 out
#define NPTS  128      // points per block
#define NTHR  256      // 8 waves of 32
#define LXP   (CIN_ + 4)   // 388: 388 % 64 == 4 -> conflict-free b64 B-tile reads
#define XSP   (C0_ + 4)    // 260: same trick for kernel 3

__device__ __forceinline__ void top3_insert(float d, int s,
                                            float& d0, float& d1, float& d2,
                                            int& i0, int& i1, int& i2) {
  if (d < d2) {
    if (d < d1) {
      d2 = d1; i2 = i1;
      if (d < d0) { d1 = d0; i1 = i0; d0 = d; i0 = s; }
      else        { d1 = d;  i1 = s; }
    } else { d2 = d; i2 = s; }
  }
}

// Async copy of one dword from global memory directly into LDS (ASYNCcnt).
__device__ __forceinline__ void async_g2lds_b32(unsigned lds_byte_off,
                                                const float* gaddr) {
  asm volatile("global_load_async_to_lds_b32 %0, %1, off"
               :
               : "v"(lds_byte_off), "v"(gaddr)
               : "memory");
}

__device__ __forceinline__ void wait_async0() {
  asm volatile("s_wait_asynccnt 0x0" ::: "memory");
}

// ---------------------------------------------------------------------------
// Kernel 1: fused 3-NN interpolation + concat + conv0 GEMM (384 -> 256).
// Block: 1 batch x 128 points. y0_pre = W0 * xcat + b0 (pre-BN) to workspace.
// ---------------------------------------------------------------------------
__global__ __launch_bounds__(NTHR) void k_interp_gemm0(
    const float* __restrict__ xyz1, const float* __restrict__ xyz2,
    const float* __restrict__ points1, const float* __restrict__ points2,
    const float* __restrict__ w0, const float* __restrict__ b0,
    float* __restrict__ y0)
{
  extern __shared__ float smem[];
  float* lx = smem;                         // [NPTS][LXP] point-major xcat tile
  float* md = lx + NPTS * LXP;              // [NTHR*3] merge scratch (dists)
  int*   mi = (int*)(md + NTHR * 3);        // [NTHR*3] merge scratch (indices)
  float* wg = (float*)(mi + NTHR * 3);      // [NPTS*3] final weights
  int*   gi = (int*)(wg + NPTS * 3);        // [NPTS*3] final indices
  const unsigned lds_dyn_base = __builtin_amdgcn_groupstaticsize();

  const int t    = threadIdx.x;
  const int b    = blockIdx.y;
  const int n0   = blockIdx.x * NPTS;
  const int p    = t & (NPTS - 1);
  const int half = t >> 7;                  // two threads share one point
  const int n    = n0 + p;

  // Warm the cache with this wave's weight rows while the scan runs.
  {
    const int lane = t & 31;
    const int m0   = ((t >> 5) * 2 + (lane >> 4)) * 16;   // wave's two tiles
    const float* row = w0 + (size_t)(m0 + (lane & 15)) * CIN_;
    #pragma unroll
    for (int off = 0; off < CIN_; off += 64)              // 256B lines
      __builtin_prefetch(row + off, 0, 1);
  }

  // ---- phase 1: 3-NN scan; xyz2 addresses are wave-uniform (scalar loads) --
  const float px = xyz1[(size_t)(b * 3 + 0) * N_ + n];
  const float py = xyz1[(size_t)(b * 3 + 1) * N_ + n];
  const float pz = xyz1[(size_t)(b * 3 + 2) * N_ + n];
  float d0 = 3e38f, d1 = 3e38f, d2 = 3e38f;
  int   i0 = 0, i1 = 0, i2 = 0;
  const float* x2 = xyz2 + (size_t)b * 3 * S_;
  const int sbeg = half * (S_ / 2), send = sbeg + (S_ / 2);
  for (int s = sbeg; s < send; ++s) {
    float dx = px - x2[s];
    float dy = py - x2[S_ + s];
    float dz = pz - x2[2 * S_ + s];
    float d  = dx * dx + dy * dy + dz * dz;
    top3_insert(d, s, d0, d1, d2, i0, i1, i2);
  }
  md[t * 3 + 0] = d0; md[t * 3 + 1] = d1; md[t * 3 + 2] = d2;
  mi[t * 3 + 0] = i0; mi[t * 3 + 1] = i1; mi[t * 3 + 2] = i2;
  __syncthreads();

  if (half == 0) {  // merge partner's top-3, compute inverse-distance weights
    #pragma unroll
    for (int k = 0; k < 3; ++k) {
      float dd = md[(t + NPTS) * 3 + k];
      int   ss = mi[(t + NPTS) * 3 + k];
      top3_insert(dd, ss, d0, d1, d2, i0, i1, i2);
    }
    float r0 = 1.f / (d0 + 1e-8f);
    float r1 = 1.f / (d1 + 1e-8f);
    float r2 = 1.f / (d2 + 1e-8f);
    float rn = 1.f / (r0 + r1 + r2);
    wg[p * 3 + 0] = r0 * rn; wg[p * 3 + 1] = r1 * rn; wg[p * 3 + 2] = r2 * rn;
    gi[p * 3 + 0] = i0;      gi[p * 3 + 1] = i1;      gi[p * 3 + 2] = i2;
  }
  __syncthreads();

  // ---- phase 2: skip concat via async global->LDS (no VGPR round trip),
  //               gather/interp (256 ch) through VALU in parallel ----------
  {
    const float* p1 = points1 + (size_t)b * D1_ * N_;
    for (int j = 0; j < D1_ / 2; ++j) {
      int c = half * (D1_ / 2) + j;
      unsigned ldsoff = lds_dyn_base + (unsigned)(p * LXP + D2_ + c) * 4u;
      async_g2lds_b32(ldsoff, p1 + (size_t)c * N_ + n);
    }

    const float wa = wg[p * 3 + 0], wb = wg[p * 3 + 1], wc = wg[p * 3 + 2];
    const int   ja = gi[p * 3 + 0], jb = gi[p * 3 + 1], jc = gi[p * 3 + 2];
    const float* p2 = points2 + (size_t)b * D2_ * S_;
    for (int j = 0; j < D2_ / 2; ++j) {
      int c = half * (D2_ / 2) + j;
      const float* row = p2 + (size_t)c * S_;
      lx[p * LXP + c] = wa * row[ja] + wb * row[jb] + wc * row[jc];
    }
  }
  wait_async0();           // this wave's async LDS writes are complete
  __syncthreads();         // all waves' ds + async writes visible

  // ---- phase 3: WMMA f32 16x16x4 GEMM: y0[m0+..][n0+..] = W0 * xcat -------
  const int lane = t & 31;
  const int wv   = t >> 5;       // wave id 0..7
  const int lrow = lane & 15;    // row within 16
  const int lhi  = lane >> 4;    // K sub-pair select (A) / same for B
  #pragma unroll
  for (int cti = 0; cti < 2; ++cti) {          // 16 ch-tiles over 8 waves
    const int m0 = (wv * 2 + cti) * 16;
    v8f acc[8];
    #pragma unroll
    for (int pt = 0; pt < 8; ++pt) acc[pt] = (v8f){0,0,0,0,0,0,0,0};
    const float* wrow = w0 + (size_t)(m0 + lrow) * CIN_ + lhi * 2;
    for (int k4 = 0; k4 < CIN_; k4 += 4) {
      v2f a = *(const v2f*)(wrow + k4);        // A tile: reused for 8 pt-tiles
      #pragma unroll
      for (int pt = 0; pt < 8; ++pt) {
        v2f bb = *(const v2f*)(lx + (pt * 16 + lrow) * LXP + k4 + lhi * 2);
        acc[pt] = __builtin_amdgcn_wmma_f32_16x16x4_f32(
            false, a, false, bb, (short)0, acc[pt], false, false);
      }
    }
    #pragma unroll
    for (int pt = 0; pt < 8; ++pt) {
      #pragma unroll
      for (int r = 0; r < 8; ++r) {
        int o  = m0 + r + 8 * lhi;
        int nn = n0 + pt * 16 + lrow;
        y0[((size_t)(b * C0_) + o) * N_ + nn] = acc[pt][r] + b0[o];
      }
    }
  }
}

// ---------------------------------------------------------------------------
// Kernel 2/4: per-channel batch stats over [B][C][N]; folds bias-included
// mean/var with gamma/beta into scale/shift. One block per channel.
// ---------------------------------------------------------------------------
__global__ __launch_bounds__(NTHR) void k_stats(
    const float* __restrict__ y, int C,
    const float* __restrict__ g, const float* __restrict__ be,
    float* __restrict__ scale, float* __restrict__ shift)
{
  __shared__ float rs[NTHR], rss[NTHR];
  const int c = blockIdx.x, t = threadIdx.x;
  float s = 0.f, ss = 0.f;
  for (int i = t; i < B_ * N_; i += NTHR) {
    int bb = i >> 13;             // N_ == 8192
    int n  = i & (N_ - 1);
    float v = y[((size_t)(bb * C) + c) * N_ + n];
    s += v; ss += v * v;
  }
  rs[t] = s; rss[t] = ss;
  __syncthreads();
  for (int off = NTHR / 2; off > 0; off >>= 1) {
    if (t < off) { rs[t] += rs[t + off]; rss[t] += rss[t + off]; }
    __syncthreads();
  }
  if (t == 0) {
    const float inv  = 1.f / (float)(B_ * N_);
    float mean = rs[0] * inv;
    float var  = rss[0] * inv - mean * mean;
    float rstd = rsqrtf(var + 1e-5f);
    float sc   = g[c] * rstd;
    scale[c] = sc;
    shift[c] = be[c] - mean * sc;
  }
}

// ---------------------------------------------------------------------------
// Kernel 3: BN0+ReLU applied while staging into LDS, then conv1 GEMM 256->128.
// ---------------------------------------------------------------------------
__global__ __launch_bounds__(NTHR) void k_gemm1(
    const float* __restrict__ y0, const float* __restrict__ scale0,
    const float* __restrict__ shift0,
    const float* __restrict__ w1, const float* __restrict__ b1,
    float* __restrict__ y1)
{
  extern __shared__ float smem[];
  float* xs = smem;                          // [NPTS][XSP]
  const int t    = threadIdx.x;
  const int b    = blockIdx.y;
  const int n0   = blockIdx.x * NPTS;
  const int p    = t & (NPTS - 1);
  const int half = t >> 7;

  for (int j = 0; j < C0_ / 2; ++j) {        // coalesced: lanes sweep n
    int c = half * (C0_ / 2) + j;
    float v = y0[((size_t)(b * C0_) + c) * N_ + n0 + p];
    v = scale0[c] * v + shift0[c];
    xs[p * XSP + c] = fmaxf(v, 0.f);
  }
  __syncthreads();

  const int lane = t & 31;
  const int wv   = t >> 5;
  const int lrow = lane & 15;
  const int lhi  = lane >> 4;
  const int m0   = wv * 16;                  // 8 ch-tiles over 8 waves
  v8f acc[8];
  #pragma unroll
  for (int pt = 0; pt < 8; ++pt) acc[pt] = (v8f){0,0,0,0,0,0,0,0};
  const float* wrow = w1 + (size_t)(m0 + lrow) * C0_ + lhi * 2;
  for (int k4 = 0; k4 < C0_; k4 += 4) {
    v2f a = *(const v2f*)(wrow + k4);
    #pragma unroll
    for (int pt = 0; pt < 8; ++pt) {
      v2f bb = *(const v2f*)(xs + (pt * 16 + lrow) * XSP + k4 + lhi * 2);
      acc[pt] = __builtin_amdgcn_wmma_f32_16x16x4_f32(
          false, a, false, bb, (short)0, acc[pt], false, false);
    }
  }
  #pragma unroll
  for (int pt = 0; pt < 8; ++pt) {
    #pragma unroll
    for (int r = 0; r < 8; ++r) {
      int o  = m0 + r + 8 * lhi;
      int nn = n0 + pt * 16 + lrow;
      y1[((size_t)(b * C1_) + o) * N_ + nn] = acc[pt][r] + b1[o];
    }
  }
}

// ---------------------------------------------------------------------------
// Kernel 5: final BN1 + ReLU elementwise -> d_out.
// ---------------------------------------------------------------------------
__global__ __launch_bounds__(NTHR) void k_bnrelu(
    const float* __restrict__ y1, const float* __restrict__ scale1,
    const float* __restrict__ shift1, float* __restrict__ out)
{
  size_t idx = (size_t)blockIdx.x * NTHR + threadIdx.x;
  const size_t total = (size_t)B_ * C1_ * N_;
  if (idx < total) {
    int c = (int)((idx / N_) % C1_);
    float v = scale1[c] * y1[idx] + shift1[c];
    out[idx] = fmaxf(v, 0.f);
  }
}

// ---------------------------------------------------------------------------
extern "C" void kernel_launch(void* const* d_in, const int* in_sizes, int n_in,
                              void* d_out, int out_size, void* d_ws, size_t ws_size,
                              hipStream_t stream)
{
  const float* xyz1    = (const float*)d_in[0];
  const float* xyz2    = (const float*)d_in[1];
  const float* points1 = (const float*)d_in[2];
  const float* points2 = (const float*)d_in[3];
  const float* w0      = (const float*)d_in[4];
  const float* b0      = (const float*)d_in[5];
  const float* g0      = (const float*)d_in[6];
  const float* be0     = (const float*)d_in[7];
  const float* w1      = (const float*)d_in[8];
  const float* b1      = (const float*)d_in[9];
  const float* g1c     = (const float*)d_in[10];
  const float* be1     = (const float*)d_in[11];

  float* ws     = (float*)d_ws;
  float* y0     = ws;                              // B*C0*N  (67 MB)
  float* y1     = y0 + (size_t)B_ * C0_ * N_;      // B*C1*N  (34 MB)
  float* scale0 = y1 + (size_t)B_ * C1_ * N_;
  float* shift0 = scale0 + C0_;
  float* scale1 = shift0 + C0_;
  float* shift1 = scale1 + C1_;

  dim3 grid(N_ / NPTS, B_);                        // (64, 8)

  size_t sh1 = (size_t)(NPTS * LXP + NTHR * 3 * 2 + NPTS * 3 * 2) * sizeof(float);
  k_interp_gemm0<<<grid, NTHR, sh1, stream>>>(xyz1, xyz2, points1, points2,
                                              w0, b0, y0);
  k_stats<<<C0_, NTHR, 0, stream>>>(y0, C0_, g0, be0, scale0, shift0);

  size_t sh2 = (size_t)(NPTS * XSP) * sizeof(float);
  k_gemm1<<<grid, NTHR, sh2, stream>>>(y0, scale0, shift0, w1, b1, y1);
  k_stats<<<C1_, NTHR, 0, stream>>>(y1, C1_, g1c, be1, scale1, shift1);

  size_t total = (size_t)B_ * C1_ * N_;
  k_bnrelu<<<(unsigned)((total + NTHR - 1) / NTHR), NTHR, 0, stream>>>(
      y1, scale1, shift1, (float*)d_out);
}